// GATModel_10986526343324
// MI455X (gfx1250) — compile-verified
//
#include <hip/hip_runtime.h>

#define NN 50000
#define EE 800000
#define ET (EE + NN)
#define DD 128
#define HH 256
#define CC 6

typedef float v2f __attribute__((ext_vector_type(2)));
typedef float v8f __attribute__((ext_vector_type(8)));

// ---------- helpers ----------
__device__ __forceinline__ unsigned enc_f(float f) {
    unsigned u = __float_as_uint(f);
    return (u & 0x80000000u) ? ~u : (u | 0x80000000u);   // order-preserving
}
__device__ __forceinline__ float dec_f(unsigned u) {
    return (u & 0x80000000u) ? __uint_as_float(u ^ 0x80000000u) : __uint_as_float(~u);
}

__global__ void zero_f(float* __restrict__ p, int n) {
    int i = blockIdx.x * blockDim.x + threadIdx.x;
    int st = gridDim.x * blockDim.x;
    for (; i < n; i += st) p[i] = 0.0f;
}

// ---------- fp32 WMMA GEMM: C[M x Ncols] = A[M x 128] * B[128 x Ncols] ----------
// 256 threads = 8 waves; block covers 16 rows x 128 cols; wave owns a 16x16 tile.
__global__ void __launch_bounds__(256)
gemm_wmma_f32(const float* __restrict__ A, const float* __restrict__ B,
              float* __restrict__ C, int Ncols,
              const float* __restrict__ bias, int relu) {
    __shared__ float sA[16 * 129];            // padded stride -> no LDS bank conflicts
    const int row0 = blockIdx.x * 16;
    const int wave = threadIdx.x >> 5;
    const int lane = threadIdx.x & 31;
    const int col0 = blockIdx.y * 128 + wave * 16;

    // cooperative stage of the 16x128 A panel
    for (int idx = threadIdx.x; idx < 16 * 128; idx += 256) {
        int r = idx >> 7;
        int c = idx & 127;
        sA[r * 129 + c] = A[(size_t)(row0 + r) * 128 + c];
    }
    __syncthreads();

    const int l  = lane & 15;
    const int kh = (lane >> 4) << 1;          // lanes 0-15: K+0/K+1, lanes 16-31: K+2/K+3
    v8f acc = {0.f, 0.f, 0.f, 0.f, 0.f, 0.f, 0.f, 0.f};

    for (int k = 0; k < 128; k += 4) {
        v2f a, b;
        a.x = sA[l * 129 + k + kh];
        a.y = sA[l * 129 + k + kh + 1];
        b.x = B[(size_t)(k + kh) * Ncols + col0 + l];
        b.y = B[(size_t)(k + kh + 1) * Ncols + col0 + l];
        acc = __builtin_amdgcn_wmma_f32_16x16x4_f32(false, a, false, b,
                                                    (short)0, acc, false, false);
    }

    const int rbase = row0 + ((lane >> 4) << 3);   // C/D layout: vgpr i = rows i / i+8
    const int col   = col0 + l;
    const float bv  = bias ? bias[col] : 0.0f;
#pragma unroll
    for (int i = 0; i < 8; ++i) {
        float v = acc[i] + bv;
        if (relu) v = fmaxf(v, 0.0f);
        C[(size_t)(rbase + i) * Ncols + col] = v;
    }
}

// ---------- per-node attention scores: ssrc = h.a_src, sdst = h.a_dst ----------
__global__ void __launch_bounds__(256)
node_scores(const float* __restrict__ h, const float* __restrict__ asrc,
            const float* __restrict__ adst, float* __restrict__ ssrc,
            float* __restrict__ sdst) {
    int wid = (blockIdx.x * blockDim.x + threadIdx.x) >> 5;
    int lane = threadIdx.x & 31;
    if (wid >= NN) return;
    const float4 hv = *(const float4*)(h + (size_t)wid * DD + lane * 4);
    const float4 a1 = *(const float4*)(asrc + lane * 4);
    const float4 a2 = *(const float4*)(adst + lane * 4);
    float s1 = hv.x * a1.x + hv.y * a1.y + hv.z * a1.z + hv.w * a1.w;
    float s2 = hv.x * a2.x + hv.y * a2.y + hv.z * a2.z + hv.w * a2.w;
#pragma unroll
    for (int off = 16; off > 0; off >>= 1) {
        s1 += __shfl_down(s1, off, 32);
        s2 += __shfl_down(s2, off, 32);
    }
    if (lane == 0) { ssrc[wid] = s1; sdst[wid] = s2; }
}

// ---------- edge pass 1: leaky-relu score + segment max ----------
__global__ void __launch_bounds__(256)
edge_score_max(const long long* __restrict__ ei, const float* __restrict__ ssrc,
               const float* __restrict__ sdst, float* __restrict__ sc,
               unsigned* __restrict__ mx) {
    int e = blockIdx.x * blockDim.x + threadIdx.x;
    if (e >= ET) return;
    long long s, d;
    if (e < EE) { s = ei[e]; d = ei[EE + e]; } else { s = d = (long long)(e - EE); }
    float v = ssrc[s] + sdst[d];
    v = v > 0.0f ? v : 0.2f * v;
    sc[e] = v;
    atomicMax(&mx[d], enc_f(v));
}

// ---------- edge pass 2: exp(score - max) + segment sum ----------
__global__ void __launch_bounds__(256)
edge_exp_sum(const long long* __restrict__ ei, float* __restrict__ sc,
             const unsigned* __restrict__ mx, float* __restrict__ ssum) {
    int e = blockIdx.x * blockDim.x + threadIdx.x;
    if (e >= ET) return;
    long long d = (e < EE) ? ei[EE + e] : (long long)(e - EE);
    float ex = __expf(sc[e] - dec_f(mx[d]));
    sc[e] = ex;
    atomicAdd(&ssum[d], ex);
}

// ---------- edge pass 3: wave-per-edge weighted scatter-add ----------
__global__ void __launch_bounds__(256)
edge_aggregate(const long long* __restrict__ ei, const float* __restrict__ sc,
               const float* __restrict__ ssum, const float* __restrict__ h,
               float* __restrict__ out) {
    int wid = (blockIdx.x * blockDim.x + threadIdx.x) >> 5;
    int lane = threadIdx.x & 31;
    if (wid >= ET) return;
    long long s, d;
    if (wid < EE) { s = ei[wid]; d = ei[EE + wid]; } else { s = d = (long long)(wid - EE); }
    float w = sc[wid] / ssum[d];
    const float4 hv = *(const float4*)(h + (size_t)s * DD + lane * 4);
    float* op = out + (size_t)d * DD + lane * 4;
    atomicAdd(op + 0, hv.x * w);
    atomicAdd(op + 1, hv.y * w);
    atomicAdd(op + 2, hv.z * w);
    atomicAdd(op + 3, hv.w * w);
}

// ---------- bias + relu over N x 128 ----------
__global__ void __launch_bounds__(256)
bias_relu(float* __restrict__ x, const float* __restrict__ b, int n) {
    int i = blockIdx.x * blockDim.x + threadIdx.x;
    if (i >= n) return;
    x[i] = fmaxf(x[i] + b[i & (DD - 1)], 0.0f);
}

// ---------- final 256 -> 6 projection, wave per node ----------
__global__ void __launch_bounds__(256)
mlp_out(const float* __restrict__ h1, const float* __restrict__ W,
        const float* __restrict__ b, float* __restrict__ out) {
    int wid = (blockIdx.x * blockDim.x + threadIdx.x) >> 5;
    int lane = threadIdx.x & 31;
    if (wid >= NN) return;
    const float* hp = h1 + (size_t)wid * HH + lane * 8;
    float acc[CC] = {0.f, 0.f, 0.f, 0.f, 0.f, 0.f};
#pragma unroll
    for (int j = 0; j < 8; ++j) {
        float hv = hp[j];
        const float* wp = W + (size_t)(lane * 8 + j) * CC;
#pragma unroll
        for (int c = 0; c < CC; ++c) acc[c] += hv * wp[c];
    }
#pragma unroll
    for (int c = 0; c < CC; ++c) {
#pragma unroll
        for (int off = 16; off > 0; off >>= 1) acc[c] += __shfl_down(acc[c], off, 32);
    }
    if (lane == 0) {
#pragma unroll
        for (int c = 0; c < CC; ++c) out[(size_t)wid * CC + c] = acc[c] + b[c];
    }
}

extern "C" void kernel_launch(void* const* d_in, const int* in_sizes, int n_in,
                              void* d_out, int out_size, void* d_ws, size_t ws_size,
                              hipStream_t stream) {
    (void)in_sizes; (void)n_in; (void)out_size; (void)ws_size;

    const float*     x   = (const float*)d_in[0];
    const long long* ei  = (const long long*)d_in[1];
    const float* W[3]  = {(const float*)d_in[2], (const float*)d_in[6],  (const float*)d_in[10]};
    const float* b[3]  = {(const float*)d_in[3], (const float*)d_in[7],  (const float*)d_in[11]};
    const float* as[3] = {(const float*)d_in[4], (const float*)d_in[8],  (const float*)d_in[12]};
    const float* ad[3] = {(const float*)d_in[5], (const float*)d_in[9],  (const float*)d_in[13]};
    const float* Wm1 = (const float*)d_in[14];
    const float* bm1 = (const float*)d_in[15];
    const float* Wm2 = (const float*)d_in[16];
    const float* bm2 = (const float*)d_in[17];

    float* ws = (float*)d_ws;
    size_t o = 0;
    float*    hA   = ws + o; o += (size_t)NN * DD;
    float*    hB   = ws + o; o += (size_t)NN * DD;
    float*    h1   = ws + o; o += (size_t)NN * HH;
    float*    ssrc = ws + o; o += NN;
    float*    sdst = ws + o; o += NN;
    unsigned* mx   = (unsigned*)(ws + o); o += NN;
    float*    ssum = ws + o; o += NN;
    float*    sc   = ws + o; o += ET;

    const dim3 blk(256);
    const float* hin = x;
    for (int L = 0; L < 3; ++L) {
        gemm_wmma_f32<<<dim3(NN / 16, 1), blk, 0, stream>>>(hin, W[L], hA, DD, nullptr, 0);
        node_scores<<<(NN + 7) / 8, blk, 0, stream>>>(hA, as[L], ad[L], ssrc, sdst);
        zero_f<<<2048, blk, 0, stream>>>((float*)mx, NN);     // enc(~-inf) == 0x0
        zero_f<<<2048, blk, 0, stream>>>(ssum, NN);
        zero_f<<<4096, blk, 0, stream>>>(hB, NN * DD);
        edge_score_max<<<(ET + 255) / 256, blk, 0, stream>>>(ei, ssrc, sdst, sc, mx);
        edge_exp_sum<<<(ET + 255) / 256, blk, 0, stream>>>(ei, sc, mx, ssum);
        edge_aggregate<<<(ET * 32 + 255) / 256, blk, 0, stream>>>(ei, sc, ssum, hA, hB);
        bias_relu<<<(NN * DD + 255) / 256, blk, 0, stream>>>(hB, b[L], NN * DD);
        hin = hB;
    }
    // MLP hidden: h1 = relu(hB @ Wm1 + bm1), Ncols = 256 -> grid.y = 2
    gemm_wmma_f32<<<dim3(NN / 16, 2), blk, 0, stream>>>(hB, Wm1, h1, HH, bm1, 1);
    // final projection 256 -> 6
    mlp_out<<<(NN + 7) / 8, blk, 0, stream>>>(h1, Wm2, bm2, (float*)d_out);
}